// FlattenDeflatten_33809982554839
// MI455X (gfx1250) — compile-verified
//
#include <hip/hip_runtime.h>
#include <hip/hip_bf16.h>

typedef float v2f __attribute__((ext_vector_type(2)));
typedef float v8f __attribute__((ext_vector_type(8)));

#define C_CH   32
#define BCH_   6
#define NPOS   57600          // 180*320
#define NBAT   32
#define TPB    (NPOS / 16)    // 3600 position-tiles per batch
#define WAVES_PER_BLOCK 8

__global__ __launch_bounds__(256)
void FlattenDeflatten_wmma_kernel(const float* __restrict__ latent,
                                  const float* __restrict__ w_flat,
                                  const float* __restrict__ b_flat,
                                  const float* __restrict__ w_deflat,
                                  const float* __restrict__ b_deflat,
                                  const float* __restrict__ embed,
                                  const int*   __restrict__ walk,
                                  float* __restrict__ recon,
                                  float* __restrict__ flat_out)
{
    const int lane = threadIdx.x & 31;
    const int wave = threadIdx.x >> 5;
    const int tile = blockIdx.x * WAVES_PER_BLOCK + wave;

    const int b  = tile / TPB;
    const int n0 = (tile - b * TPB) * 16;

    const int p    = lane & 15;   // position-in-tile (B/C cols) and M-row (A)
    const int half = lane >> 4;   // lane half selects K pair / M row group

    const int n = n0 + p;         // walk-order position
    const int r = walk[n];        // raster position

    // ---------- A operand for GEMM1 (W_flat, rows 6..15 padded) ------------
    // Clamped address keeps every lane's load valid. No zero-masking needed:
    // accumulator rows 6..15 compute finite junk (row-0 weights) that is
    // never stored and never sourced by the ds_bpermute rearrange below.
    const int pm = (p < BCH_) ? p : 0;
    const float* wfrow = w_flat + pm * C_CH + 2 * half;

    float af[8][2];
#pragma unroll
    for (int kk = 0; kk < 8; ++kk) {
        af[kk][0] = wfrow[4 * kk];
        af[kk][1] = wfrow[4 * kk + 1];
    }

    // ---------- gather B operand: X[channel, walk[n]] -----------------------
    const float* xbase = latent + (size_t)b * C_CH * NPOS
                       + (size_t)(2 * half) * NPOS + r;
    float bx[8][2];
#pragma unroll
    for (int kk = 0; kk < 8; ++kk) {
        bx[kk][0] = xbase[(size_t)(4 * kk)     * NPOS];
        bx[kk][1] = xbase[(size_t)(4 * kk + 1) * NPOS];
    }

    // ---------------- GEMM1: flat(6x16) = W_flat(6x32) * X(32x16) ----------
    v8f c1 = {};
#pragma unroll
    for (int kk = 0; kk < 8; ++kk) {
        v2f a, bm;
        a.x  = af[kk][0]; a.y  = af[kk][1];
        bm.x = bx[kk][0]; bm.y = bx[kk][1];
        c1 = __builtin_amdgcn_wmma_f32_16x16x4_f32(
                 false, a, false, bm, (short)0, c1, false, false);
    }

    // add flatten bias to the 6 valid output rows
    c1[0] += b_flat[0]; c1[1] += b_flat[1]; c1[2] += b_flat[2];
    c1[3] += b_flat[3]; c1[4] += b_flat[4]; c1[5] += b_flat[5];

    // -------- rearrange flat tile from C/D layout into B layout ------------
    // B needs f[K, pos]: lower lanes K={0,1,4,5}, upper lanes K={2,3,(6,7)=0}.
    // Source: f[K, pos] sits in lane `pos` (0..15) of c1[K].
    const float t0 = __shfl(c1[0], p, 32);
    const float t1 = __shfl(c1[1], p, 32);
    const float t2 = __shfl(c1[2], p, 32);
    const float t3 = __shfl(c1[3], p, 32);
    const float t4 = __shfl(c1[4], p, 32);
    const float t5 = __shfl(c1[5], p, 32);

    v2f b0, b1;                       // K-chunk 0 (K=0..3) and 1 (K=4..7)
    b0.x = half ? t2   : t0;
    b0.y = half ? t3   : t1;
    b1.x = half ? 0.0f : t4;          // B rows K=6,7 are zero padding;
    b1.y = half ? 0.0f : t5;          // that zero makes A-side K=6,7 don't-care

    // ---- A operand for GEMM2 (W_deflat 32x6, K padded to 8) ---------------
    // Addresses valid for every lane; K=6,7 contributions are killed by the
    // zeroed B rows above, so no masking of A is needed.
    const int k0a = 2 * half;         // kk=0: K = {2h, 2h+1}, always < 6
    v2f a00, a01, a10, a11;
    {
        const float* wd0 = w_deflat + p * BCH_;          // rows 0..15
        const float* wd1 = w_deflat + (16 + p) * BCH_;   // rows 16..31
        a00.x = wd0[k0a]; a00.y = wd0[k0a + 1];
        a01.x = wd0[4];   a01.y = wd0[5];                // K={4,5} (B=0 if K>=6)
        a10.x = wd1[k0a]; a10.y = wd1[k0a + 1];
        a11.x = wd1[4];   a11.y = wd1[5];
    }

    // ---------------- GEMM2: recon(32x16) = W_deflat(32x6) * flat(6x16) ----
    v8f c2a = {}, c2b = {};
    c2a = __builtin_amdgcn_wmma_f32_16x16x4_f32(
              false, a00, false, b0, (short)0, c2a, false, false);
    c2a = __builtin_amdgcn_wmma_f32_16x16x4_f32(
              false, a01, false, b1, (short)0, c2a, false, false);
    c2b = __builtin_amdgcn_wmma_f32_16x16x4_f32(
              false, a10, false, b0, (short)0, c2b, false, false);
    c2b = __builtin_amdgcn_wmma_f32_16x16x4_f32(
              false, a11, false, b1, (short)0, c2b, false, false);

    // store `flat` output: row M=v lives in lanes 0..15 of vgpr v
    // (after all WMMAs so EXEC is all-1s for every matrix op)
    if (half == 0) {
        float* fo = flat_out + ((size_t)b * BCH_) * NPOS + n;
#pragma unroll
        for (int v = 0; v < BCH_; ++v)
            fo[(size_t)v * NPOS] = c1[v];
    }

    // -------- epilogue: + b_deflat + spatial_embed, scatter via walk -------
    float* rbase = recon + (size_t)b * C_CH * NPOS;
#pragma unroll
    for (int v = 0; v < 8; ++v) {
        const int m0 = v + 8 * half;          // rows 0..15  (tile a)
        const int m1 = m0 + 16;               // rows 16..31 (tile b)
        const float ya = c2a[v] + b_deflat[m0] + embed[(size_t)m0 * NPOS + n];
        const float yb = c2b[v] + b_deflat[m1] + embed[(size_t)m1 * NPOS + n];
        rbase[(size_t)m0 * NPOS + r] = ya;
        rbase[(size_t)m1 * NPOS + r] = yb;
    }
}

extern "C" void kernel_launch(void* const* d_in, const int* in_sizes, int n_in,
                              void* d_out, int out_size, void* d_ws, size_t ws_size,
                              hipStream_t stream)
{
    (void)in_sizes; (void)n_in; (void)out_size; (void)d_ws; (void)ws_size;

    const float* latent   = (const float*)d_in[0];
    const float* w_flat   = (const float*)d_in[1];
    const float* b_flat   = (const float*)d_in[2];
    const float* w_deflat = (const float*)d_in[3];
    const float* b_deflat = (const float*)d_in[4];
    const float* embed    = (const float*)d_in[5];
    const int*   walk     = (const int*)  d_in[6];
    // d_in[7] = unwalk_idx, not needed (we scatter through walk directly)

    float* recon    = (float*)d_out;                            // B*C*N floats
    float* flat_out = (float*)d_out + (size_t)NBAT * C_CH * NPOS;

    const int tiles  = NBAT * TPB;                 // 115200 wave-tiles
    const int blocks = tiles / WAVES_PER_BLOCK;    // 14400

    FlattenDeflatten_wmma_kernel<<<blocks, 256, 0, stream>>>(
        latent, w_flat, b_flat, w_deflat, b_deflat, embed, walk,
        recon, flat_out);
}